// SpikeEncoder_50697793962790
// MI455X (gfx1250) — compile-verified
//
#include <hip/hip_runtime.h>
#include <hip/hip_bf16.h>

// SpikeEncoder for MI455X (gfx1250): store-bandwidth-bound one-hot scatter.
// ~21 MB input read (L2-resident between passes), ~629 MB streaming output
// written with non-temporal B128 stores. No matmul content -> no WMMA; the
// CDNA5-relevant paths here are TH=NT cache hints, global_prefetch_b8, and
// wave32 shuffle reductions.

#define T_STEPS 30

typedef float v4f __attribute__((ext_vector_type(4)));

// Monotone mapping float -> uint so integer atomicMin/atomicMax implement
// exact float min/max for any sign.
__device__ __forceinline__ unsigned orderKey(float f) {
    unsigned b = __float_as_uint(f);
    return (b & 0x80000000u) ? ~b : (b | 0x80000000u);
}
__device__ __forceinline__ float keyToFloat(unsigned k) {
    unsigned b = (k & 0x80000000u) ? (k & 0x7FFFFFFFu) : ~k;
    return __uint_as_float(b);
}

__global__ void spike_init_ws(unsigned* __restrict__ ws) {
    // ws[0]: min-key (init to +inf key), ws[1]: max-key (init to -inf key).
    ws[0] = 0xFFFFFFFFu;
    ws[1] = 0x00000000u;
}

__global__ void spike_minmax(const float* __restrict__ in,
                             unsigned* __restrict__ ws, int n4) {
    const v4f* __restrict__ in4 = (const v4f*)in;
    const int tid    = blockIdx.x * blockDim.x + threadIdx.x;
    const int stride = gridDim.x * blockDim.x;

    float mn =  __builtin_inff();
    float mx = -__builtin_inff();
    for (int i = tid; i < n4; i += stride) {
        // Prefetch next iteration's line into the cache hierarchy.
        __builtin_prefetch(in4 + i + stride, 0, 0);
        v4f v = in4[i];  // regular-temporal: warm L2 for the encode pass
        mn = fminf(mn, fminf(fminf(v.x, v.y), fminf(v.z, v.w)));
        mx = fmaxf(mx, fmaxf(fmaxf(v.x, v.y), fmaxf(v.z, v.w)));
    }

    // wave32 butterfly reduction (gfx1250 is wave32-only).
    #pragma unroll
    for (int off = 16; off > 0; off >>= 1) {
        mn = fminf(mn, __shfl_xor(mn, off, 32));
        mx = fmaxf(mx, __shfl_xor(mx, off, 32));
    }

    __shared__ float smn[8];  // 256 threads = 8 wave32s
    __shared__ float smx[8];
    const int lane = threadIdx.x & 31;
    const int wave = threadIdx.x >> 5;
    if (lane == 0) { smn[wave] = mn; smx[wave] = mx; }
    __syncthreads();

    if (wave == 0) {
        mn = (lane < 8) ? smn[lane] :  __builtin_inff();
        mx = (lane < 8) ? smx[lane] : -__builtin_inff();
        #pragma unroll
        for (int off = 4; off > 0; off >>= 1) {
            mn = fminf(mn, __shfl_xor(mn, off, 32));
            mx = fmaxf(mx, __shfl_xor(mx, off, 32));
        }
        if (lane == 0) {
            atomicMin(&ws[0], orderKey(mn));
            atomicMax(&ws[1], orderKey(mx));
        }
    }
}

__global__ void spike_encode(const float* __restrict__ in,
                             float* __restrict__ out,
                             const unsigned* __restrict__ ws,
                             int n4, int n) {
    const int i = blockIdx.x * blockDim.x + threadIdx.x;
    if (i >= n4) return;

    const float tmin  = keyToFloat(ws[0]);   // uniform scalar loads
    const float tmax  = keyToFloat(ws[1]);
    const float denom = tmax - tmin + 1e-8f;

    // Last-use read of the input (already L2-resident from pass 1).
    v4f v = __builtin_nontemporal_load((const v4f*)in + i);

    int lat[4];
    #pragma unroll
    for (int k = 0; k < 4; ++k) {
        // Exact reference math: norm = (x - tmin)/(tmax - tmin + 1e-8)
        //                       lat  = floor((1 - norm) * (T-1))
        float norm = (v[k] - tmin) / denom;
        int l = (int)floorf((1.0f - norm) * (float)(T_STEPS - 1));
        l = l < 0 ? 0 : (l > T_STEPS - 1 ? T_STEPS - 1 : l);
        lat[k] = l;
    }

    const size_t base = (size_t)i * 4;
    #pragma unroll
    for (int t = 0; t < T_STEPS; ++t) {
        v4f o;
        o.x = (lat[0] == t) ? 1.0f : 0.0f;
        o.y = (lat[1] == t) ? 1.0f : 0.0f;
        o.z = (lat[2] == t) ? 1.0f : 0.0f;
        o.w = (lat[3] == t) ? 1.0f : 0.0f;
        // Streaming 629 MB: non-temporal B128 stores, fully coalesced
        // (512 B per wave per instruction), keep L2 for the input.
        __builtin_nontemporal_store(o, (v4f*)(out + (size_t)t * (size_t)n + base));
    }
}

extern "C" void kernel_launch(void* const* d_in, const int* in_sizes, int n_in,
                              void* d_out, int out_size, void* d_ws, size_t ws_size,
                              hipStream_t stream) {
    const float* topo = (const float*)d_in[0];
    float* out        = (float*)d_out;
    unsigned* ws      = (unsigned*)d_ws;

    const int n  = in_sizes[0];   // 5 * 1024 * 1024 (divisible by 4)
    const int n4 = n >> 2;

    // 1) re-seed min/max cells every call (ws persists across graph replays)
    spike_init_ws<<<1, 1, 0, stream>>>(ws);

    // 2) global min/max: 1024 blocks x 256 threads, grid-stride over float4s
    spike_minmax<<<1024, 256, 0, stream>>>(topo, ws, n4);

    // 3) one-hot encode: one float4 per thread, 30 NT float4 stores each
    const int eblocks = (n4 + 255) / 256;
    spike_encode<<<eblocks, 256, 0, stream>>>(topo, out, ws, n4, n);
}